// ExperimentModel_36120674959535
// MI455X (gfx1250) — compile-verified
//
#include <hip/hip_runtime.h>
#include <cstddef>

// Problem constants (fixed by the reference).
#define BB 128          // batch
#define TT 512          // timesteps
#define HH 512          // hidden
#define VV 96           // vocab
#define GG 2048         // 4*HH gate width
#define MM 16           // batch rows per workgroup (one WMMA tile row)
#define NTHREADS 512    // 16 wave32 waves; 16 waves * 8 N-tiles = 128 tiles = N/16

typedef float v2f __attribute__((ext_vector_type(2)));
typedef float v8f __attribute__((ext_vector_type(8)));

// ---- fast transcendentals -------------------------------------------------
// CDNA5 has a hardware v_tanh_f32 TRANS op; use it if the builtin exists.
#if __has_builtin(__builtin_amdgcn_tanhf)
__device__ __forceinline__ float fast_tanh(float x) {
    return __builtin_amdgcn_tanhf(x);
}
#else
// Branch-free fallback: tanh(x) = sign(x) * (1 - 2/(1+exp(2|x|)))
__device__ __forceinline__ float fast_tanh(float x) {
    const float ax = __builtin_fabsf(x);
    const float e  = __expf(-2.0f * ax);
    const float t  = (1.0f - e) * __builtin_amdgcn_rcpf(1.0f + e);
    return __builtin_copysignf(t, x);
}
#endif

// sigmoid(x) = 0.5*tanh(x/2) + 0.5  (mul + tanh + fma, branch-free)
__device__ __forceinline__ float sigm(float x) {
    return __builtin_fmaf(0.5f, fast_tanh(0.5f * x), 0.5f);
}

// D = A(16x4, from LDS) * B(4x16, from L2-resident weights) + C, f32 WMMA.
// A layout (ISA 7.12.2, 32-bit A 16x4): lane<16 holds K=k0,k0+1 ; lane>=16 holds K=k0+2,k0+3 ; M=lane%16
// B layout (4x16, row striped across lanes): lane<16 -> K=k0,k0+1 at N=lane ; lane>=16 -> K=k0+2,k0+3
// C/D layout: vgpr r -> M = r + 8*(lane/16), N = lane%16 within tile.
__device__ __forceinline__ v8f wmma_f32x4(v2f a, v2f b, v8f c) {
    return __builtin_amdgcn_wmma_f32_16x16x4_f32(false, a, false, b, (short)0, c,
                                                 false, false);
}

// One K=512 half of an LSTM-layer GEMM over 8 gate-aligned N-tiles.
// X: LDS activations [MM][HH] (row-major). W: row-major [*, GG], rows wrow0..wrow0+511.
__device__ __forceinline__ void gemm_half_lstm(v8f acc[8],
                                               const float* __restrict__ W, int wrow0,
                                               const float* __restrict__ X,
                                               int wave, int lane) {
    const int mrow = lane & 15;
    const int lh   = lane >> 4;
    const int col  = lane & 15;
    const float* __restrict__ Wbase = W + (size_t)wrow0 * GG + col;
#pragma unroll 1
    for (int k0 = 0; k0 < HH; k0 += 4) {
        const int kA = k0 + 2 * lh;
        v2f a;
        a.x = X[mrow * HH + kA];        // ds_load_2addr_b32 (contiguous, 8B aligned)
        a.y = X[mrow * HH + kA + 1];
        const float* __restrict__ wk = Wbase + (size_t)kA * GG;
#pragma unroll
        for (int t = 0; t < 8; ++t) {
            const int g  = t >> 1;
            const int hf = t & 1;
            const int n0 = g * HH + wave * 32 + hf * 16;
            v2f b;
            b.x = wk[n0];
            b.y = wk[n0 + GG];
            acc[t] = wmma_f32x4(a, b, acc[t]);
        }
        if (k0 + 8 <= HH)               // global_prefetch_b8: next K-step weight rows
            __builtin_prefetch((const void*)(wk + 4 * GG), 0, 3);
    }
}

// Gate nonlinearity + cell update, entirely per-lane (gate-aligned tiling puts
// i/j/f/o for the same (m,h) into the same lane), then publish h to LDS.
__device__ __forceinline__ void lstm_combine(v8f acc[8], float c[2][8],
                                             float* __restrict__ Hdst,
                                             int wave, int lane) {
    const int lh  = lane >> 4;
    const int col = lane & 15;
    float hn[2][8];
#pragma unroll
    for (int hf = 0; hf < 2; ++hf) {
#pragma unroll
        for (int r = 0; r < 8; ++r) {
            const float zi = acc[0 + hf][r];
            const float zj = acc[2 + hf][r];
            const float zf = acc[4 + hf][r];
            const float zo = acc[6 + hf][r];
            const float cn = c[hf][r] * sigm(zf + 1.0f) + sigm(zi) * fast_tanh(zj);
            c[hf][r]  = cn;
            hn[hf][r] = fast_tanh(cn) * sigm(zo);
        }
    }
    __syncthreads();  // all waves done reading old h before overwrite
#pragma unroll
    for (int hf = 0; hf < 2; ++hf)
#pragma unroll
        for (int r = 0; r < 8; ++r)
            Hdst[(r + 8 * lh) * HH + wave * 32 + hf * 16 + col] = hn[hf][r];
    __syncthreads();  // new h visible before next layer reads
}

extern "C" __global__ __launch_bounds__(NTHREADS, 1)
void lstm3_char_kernel(const int* __restrict__ ch,
                       const float* __restrict__ W0, const float* __restrict__ b0,
                       const float* __restrict__ W1, const float* __restrict__ b1,
                       const float* __restrict__ W2, const float* __restrict__ b2,
                       const float* __restrict__ Wout, const float* __restrict__ bout,
                       float* __restrict__ logits_out, float* __restrict__ probs_out,
                       float* __restrict__ amax_out) {
    extern __shared__ float smem[];
    float* Hs0 = smem;                 // [MM][HH] layer-0 h
    float* Hs1 = Hs0 + MM * HH;       // [MM][HH] layer-1 h
    float* Hs2 = Hs1 + MM * HH;       // [MM][HH] layer-2 h
    float* Lg  = Hs2 + MM * HH;       // [MM][VV] logits staging

    const int tid  = threadIdx.x;
    const int lane = tid & 31;
    const int wave = tid >> 5;
    const int wg   = blockIdx.x;      // owns batch rows [wg*16, wg*16+16)
    const int lh   = lane >> 4;
    const int col  = lane & 15;

    for (int i = tid; i < 3 * MM * HH; i += NTHREADS) smem[i] = 0.0f;

    // Cell state in VGPRs: static (wave,half,r,lane) -> (m,h) mapping across steps.
    float c0[2][8] = {}, c1[2][8] = {}, c2[2][8] = {};
    __syncthreads();

    for (int ts = 0; ts < TT; ++ts) {
        // ---------------- layer 0 ----------------
        int cidx[8];
#pragma unroll
        for (int r = 0; r < 8; ++r)
            cidx[r] = ch[(size_t)(wg * MM + r + 8 * lh) * TT + ts];

        v8f acc[8];
#pragma unroll
        for (int t = 0; t < 8; ++t) {
            const int g  = t >> 1;
            const int hf = t & 1;
            const int n  = g * HH + wave * 32 + hf * 16 + col;
            const float bb = b0[n];
#pragma unroll
            for (int r = 0; r < 8; ++r)   // one-hot GEMM == row gather of W0[:96]
                acc[t][r] = bb + W0[(size_t)cidx[r] * GG + n];
        }
        gemm_half_lstm(acc, W0, VV, Hs0, wave, lane);   // h-part: W0 rows 96..607
        lstm_combine(acc, c0, Hs0, wave, lane);

        // ---------------- layer 1 ----------------
#pragma unroll
        for (int t = 0; t < 8; ++t) {
            const int g  = t >> 1;
            const int hf = t & 1;
            const float bb = b1[g * HH + wave * 32 + hf * 16 + col];
#pragma unroll
            for (int r = 0; r < 8; ++r) acc[t][r] = bb;
        }
        gemm_half_lstm(acc, W1, 0, Hs0, wave, lane);    // x = h0(t),  W1 rows 0..511
        gemm_half_lstm(acc, W1, HH, Hs1, wave, lane);   // h = h1(t-1),W1 rows 512..1023
        lstm_combine(acc, c1, Hs1, wave, lane);

        // ---------------- layer 2 ----------------
#pragma unroll
        for (int t = 0; t < 8; ++t) {
            const int g  = t >> 1;
            const int hf = t & 1;
            const float bb = b2[g * HH + wave * 32 + hf * 16 + col];
#pragma unroll
            for (int r = 0; r < 8; ++r) acc[t][r] = bb;
        }
        gemm_half_lstm(acc, W2, 0, Hs1, wave, lane);
        gemm_half_lstm(acc, W2, HH, Hs2, wave, lane);
        lstm_combine(acc, c2, Hs2, wave, lane);

        // ------------- output projection (N=96 -> 6 tiles on waves 0..5) -------------
        if (wave < 6) {
            v8f ap;
            const float bb = bout[wave * 16 + col];
#pragma unroll
            for (int r = 0; r < 8; ++r) ap[r] = bb;
            const float* __restrict__ Wb = Wout + wave * 16 + col;
#pragma unroll 1
            for (int k0 = 0; k0 < HH; k0 += 4) {
                const int kA = k0 + 2 * lh;
                v2f a;
                a.x = Hs2[(lane & 15) * HH + kA];
                a.y = Hs2[(lane & 15) * HH + kA + 1];
                v2f b;
                b.x = Wb[(size_t)kA * VV];
                b.y = Wb[(size_t)(kA + 1) * VV];
                ap = wmma_f32x4(a, b, ap);
            }
#pragma unroll
            for (int r = 0; r < 8; ++r)
                Lg[(r + 8 * lh) * VV + wave * 16 + col] = ap[r];
        }
        __syncthreads();

        // ------------- softmax + argmax, one thread per batch row -------------
        if (tid < MM) {
            const float* __restrict__ row = Lg + tid * VV;
            float mx = row[0];
            for (int v = 1; v < VV; ++v) mx = fmaxf(mx, row[v]);
            float s = 0.0f;
            for (int v = 0; v < VV; ++v) s += __expf(row[v] - mx);
            const float inv = __builtin_amdgcn_rcpf(s);
            const size_t ob = ((size_t)(wg * MM + tid) * TT + ts) * VV;
            int am = 0;
            float best = row[0];
            for (int v = 0; v < VV; ++v) {
                const float l = row[v];
                logits_out[ob + v] = l;
                probs_out[ob + v]  = __expf(l - mx) * inv;
                if (l > best) { best = l; am = v; }
            }
            amax_out[(size_t)(wg * MM + tid) * TT + ts] = (float)am;
        }
        __syncthreads();  // protect Lg before next step's projection writes
    }
}

extern "C" void kernel_launch(void* const* d_in, const int* in_sizes, int n_in,
                              void* d_out, int out_size, void* d_ws, size_t ws_size,
                              hipStream_t stream) {
    (void)in_sizes; (void)n_in; (void)out_size; (void)d_ws; (void)ws_size;
    const int*   ch   = (const int*)d_in[0];
    const float* W0   = (const float*)d_in[1];
    const float* b0   = (const float*)d_in[2];
    const float* W1   = (const float*)d_in[3];
    const float* b1   = (const float*)d_in[4];
    const float* W2   = (const float*)d_in[5];
    const float* b2   = (const float*)d_in[6];
    const float* Wout = (const float*)d_in[7];
    const float* bout = (const float*)d_in[8];

    float* out    = (float*)d_out;
    float* logits = out;
    float* probs  = out + (size_t)BB * TT * VV;
    float* amax   = out + (size_t)2 * BB * TT * VV;

    const size_t shmem = (size_t)(3 * MM * HH + MM * VV) * sizeof(float);  // ~104 KB < 320 KB LDS
    (void)hipFuncSetAttribute((const void*)lstm3_char_kernel,
                              hipFuncAttributeMaxDynamicSharedMemorySize, (int)shmem);

    hipLaunchKernelGGL(lstm3_char_kernel, dim3(BB / MM), dim3(NTHREADS), shmem, stream,
                       ch, W0, b0, W1, b1, W2, b2, Wout, bout, logits, probs, amax);
}